// KMeans_47648367182540
// MI455X (gfx1250) — compile-verified
//
#include <hip/hip_runtime.h>

// KMeans (N=131072, D=64, K=512, 5 iters) for MI455X / gfx1250 (wave32, WMMA).
// Distance GEMM uses V_WMMA_F32_16X16X4_F32 (the fp32 WMMA shape) so results
// match the fp32 reference. X (32MB) is L2-resident; centers (128KB) stream
// from L2 every pass. Scatter-mean via global fp32 atomics into d_ws.
//
// Round-2 changes: B fragments for a whole center tile are batch-loaded into
// registers (16 independent global_load_b64 -> 16-deep MLP instead of a
// s_wait_loadcnt 0x0 stall per load), and each tile uses 4 independent WMMA
// accumulation chains (2 M-tiles x 2 K-halves) to cover WMMA RAW latency.

#define N_PTS 131072
#define DIM   64
#define KC    512
#define MITER 5

typedef float v2f __attribute__((ext_vector_type(2)));
typedef float v8f __attribute__((ext_vector_type(8)));

// ---------------------------------------------------------------------------
// Assignment + scatter-accumulate kernel.
// Block = 256 threads = 8 waves; each wave handles 32 rows (two 16-row tiles).
// Grid = N/256 = 512 blocks.
// ---------------------------------------------------------------------------
__global__ void __launch_bounds__(256)
kmeans_assign(const float* __restrict__ X, const float* __restrict__ C,
              int* __restrict__ assign_out, float* __restrict__ sums,
              float* __restrict__ counts)
{
    __shared__ float lds_c2[KC];

    const int tid = threadIdx.x;

    // Per-block center squared norms (centers are hot in L2; 2 per thread).
    for (int k = tid; k < KC; k += 256) {
        const float* c = C + (size_t)k * DIM;
        float s = 0.f;
        #pragma unroll
        for (int d = 0; d < DIM; ++d) s = fmaf(c[d], c[d], s);
        lds_c2[k] = s;
    }
    __syncthreads();

    const int lane = tid & 31;
    const int wave = tid >> 5;
    const int lrow = lane & 15;          // A row / B col within the 16-tile
    const int dofs = (lane >> 4) << 1;   // K-offset within 4-chunk: 0 or 2
    const int row_base = (blockIdx.x * 8 + wave) * 32;

    // Preload A fragments: per lane a float2 of X[row][d_chunk + dofs .. +1]
    // (ISA 32-bit A 16x4 layout: lanes 0-15 -> K=0,1 ; lanes 16-31 -> K=2,3).
    v2f A[2][16];
    #pragma unroll
    for (int t = 0; t < 2; ++t) {
        const float* xr = X + (size_t)(row_base + t * 16 + lrow) * DIM + dofs;
        #pragma unroll
        for (int c = 0; c < 16; ++c)
            A[t][c] = *(const v2f*)(xr + c * 4);
    }

    float bv[2][8];
    int   bi[2][8];
    #pragma unroll
    for (int t = 0; t < 2; ++t)
        #pragma unroll
        for (int r = 0; r < 8; ++r) { bv[t][r] = 3.4e38f; bi[t][r] = 0; }

    // Sweep all 32 center tiles (16 centers each).
    for (int ct = 0; ct < KC / 16; ++ct) {
        const int c_base = ct * 16;
        const float* cr = C + (size_t)(c_base + lrow) * DIM + dofs;

        // Prefetch next tile first so it overlaps this tile's compute.
        if (ct + 1 < KC / 16)
            __builtin_prefetch(cr + 16 * DIM, 0, 1);   // global_prefetch_b8

        // Batch-load all 16 B fragments (independent b64 loads, deep MLP).
        v2f Bf[16];
        #pragma unroll
        for (int c = 0; c < 16; ++c)
            Bf[c] = *(const v2f*)(cr + c * 4);

        // 4 independent WMMA chains: 2 M-tiles x 2 K-halves (8 deep each).
        v8f acc00 = {}, acc01 = {}, acc10 = {}, acc11 = {};
        #pragma unroll
        for (int c = 0; c < 8; ++c) {
            acc00 = __builtin_amdgcn_wmma_f32_16x16x4_f32(
                        false, A[0][c], false, Bf[c], (short)0, acc00, false, false);
            acc10 = __builtin_amdgcn_wmma_f32_16x16x4_f32(
                        false, A[1][c], false, Bf[c], (short)0, acc10, false, false);
            acc01 = __builtin_amdgcn_wmma_f32_16x16x4_f32(
                        false, A[0][c + 8], false, Bf[c + 8], (short)0, acc01, false, false);
            acc11 = __builtin_amdgcn_wmma_f32_16x16x4_f32(
                        false, A[1][c + 8], false, Bf[c + 8], (short)0, acc11, false, false);
        }
        const v8f dot0 = acc00 + acc01;
        const v8f dot1 = acc10 + acc11;

        // score = c^2 - 2*dot  (x^2 is row-constant: irrelevant for argmin)
        const float c2v  = lds_c2[c_base + lrow];
        const int   kidx = c_base + lrow;
        #pragma unroll
        for (int r = 0; r < 8; ++r) {
            float s0 = fmaf(-2.f, dot0[r], c2v);
            if (s0 < bv[0][r]) { bv[0][r] = s0; bi[0][r] = kidx; }
            float s1 = fmaf(-2.f, dot1[r], c2v);
            if (s1 < bv[1][r]) { bv[1][r] = s1; bi[1][r] = kidx; }
        }
    }

    // Cross-lane argmin inside each 16-lane half (wave32 shuffles).
    // C/D layout: lanes 0-15 hold rows M=r, lanes 16-31 hold rows M=8+r.
    #pragma unroll
    for (int t = 0; t < 2; ++t) {
        #pragma unroll
        for (int r = 0; r < 8; ++r) {
            float v = bv[t][r];
            int   i = bi[t][r];
            #pragma unroll
            for (int m = 1; m < 16; m <<= 1) {
                float ov = __shfl_xor(v, m, 32);
                int   oi = __shfl_xor(i, m, 32);
                if (ov < v || (ov == v && oi < i)) { v = ov; i = oi; }
            }
            bv[t][r] = v; bi[t][r] = i;
        }
    }

    // Lanes 0 and 16 hold the winners; write assignments (int32 bit pattern).
    if (lrow == 0) {
        const int mo = (lane >> 4) * 8;
        #pragma unroll
        for (int t = 0; t < 2; ++t)
            #pragma unroll
            for (int r = 0; r < 8; ++r)
                assign_out[row_base + t * 16 + mo + r] = bi[t][r];
    }

    // Scatter-accumulate: broadcast k per row, all 32 lanes add 2 floats each.
    #pragma unroll
    for (int t = 0; t < 2; ++t) {
        #pragma unroll
        for (int m = 0; m < 16; ++m) {
            const int k   = __shfl(bi[t][m & 7], (m < 8) ? 0 : 16, 32);
            const int row = row_base + t * 16 + m;
            const float* xr = X + (size_t)row * DIM;
            atomicAdd(&sums[(size_t)k * DIM + lane * 2 + 0], xr[lane * 2 + 0]);
            atomicAdd(&sums[(size_t)k * DIM + lane * 2 + 1], xr[lane * 2 + 1]);
            if (lane == 0) atomicAdd(&counts[k], 1.0f);
        }
    }
}

// ---------------------------------------------------------------------------
__global__ void __launch_bounds__(256)
kmeans_zero(float* __restrict__ sums, float* __restrict__ counts)
{
    const int i = blockIdx.x * 256 + threadIdx.x;
    if (i < KC * DIM) sums[i] = 0.f;
    if (i < KC)       counts[i] = 0.f;
}

// new_centers = sums / max(counts,1); zero entries re-seeded from X[repl[k]].
__global__ void __launch_bounds__(256)
kmeans_update(const float* __restrict__ sums, const float* __restrict__ counts,
              const float* __restrict__ X, const int* __restrict__ repl,
              float* __restrict__ C)
{
    const int i = blockIdx.x * 256 + threadIdx.x;   // i in [0, K*D)
    const int k = i >> 6;
    const int d = i & 63;
    float v = sums[i] / fmaxf(counts[k], 1.0f);
    if (v == 0.0f) v = X[(size_t)repl[k] * DIM + d];
    C[i] = v;
}

// ---------------------------------------------------------------------------
extern "C" void kernel_launch(void* const* d_in, const int* in_sizes, int n_in,
                              void* d_out, int out_size, void* d_ws, size_t ws_size,
                              hipStream_t stream)
{
    (void)in_sizes; (void)n_in; (void)out_size; (void)ws_size;

    const float* X        = (const float*)d_in[0];   // [N, D] fp32
    const float* centers0 = (const float*)d_in[1];   // [K, D] fp32
    const int*   repl     = (const int*)d_in[2];     // [MITER, K] int32

    float* ws     = (float*)d_ws;
    float* Ccur   = ws;                    // [K*D]
    float* sums   = ws + KC * DIM;         // [K*D]
    float* counts = ws + 2 * KC * DIM;     // [K]

    int*   assign_out  = (int*)d_out;              // first N elems: int32 bits
    float* centers_out = (float*)d_out + N_PTS;    // then K*D fp32 centers

    // Work on a private copy of centers (inputs must not be mutated).
    hipMemcpyAsync(Ccur, centers0, (size_t)KC * DIM * sizeof(float),
                   hipMemcpyDeviceToDevice, stream);

    for (int it = 0; it < MITER; ++it) {
        kmeans_zero<<<(KC * DIM + 255) / 256, 256, 0, stream>>>(sums, counts);
        kmeans_assign<<<N_PTS / 256, 256, 0, stream>>>(X, Ccur, assign_out,
                                                       sums, counts);
        kmeans_update<<<(KC * DIM) / 256, 256, 0, stream>>>(sums, counts, X,
                                                            repl + it * KC, Ccur);
    }

    hipMemcpyAsync(centers_out, Ccur, (size_t)KC * DIM * sizeof(float),
                   hipMemcpyDeviceToDevice, stream);
}